// AttentiveMatchingLayer_69939247448444
// MI455X (gfx1250) — compile-verified
//
#include <hip/hip_runtime.h>

// Shapes from the reference
#define B 32
#define T 256
#define D 512
#define P 20
#define EPS 1e-12f

typedef float v2f __attribute__((ext_vector_type(2)));
typedef float v8f __attribute__((ext_vector_type(8)));

__device__ __forceinline__ float waveReduceSum(float v) {
    #pragma unroll
    for (int off = 16; off > 0; off >>= 1)
        v += __shfl_xor(v, off, 32);
    return v;
}

// ---------------------------------------------------------------------------
// Kernel 1: row L2-normalize inp_a -> na, inp_b -> nb.  One wave per row of D=512.
// Lane handles 16 consecutive floats (4x float4, coalesced b128 loads).
// ---------------------------------------------------------------------------
__global__ __launch_bounds__(256) void normalize_rows(const float* __restrict__ a,
                                                      const float* __restrict__ b,
                                                      float* __restrict__ na,
                                                      float* __restrict__ nb) {
    const int wave = threadIdx.x >> 5;
    const int lane = threadIdx.x & 31;
    const int row  = blockIdx.x * 8 + wave;           // 0 .. 2*B*T-1
    const bool isB = row >= B * T;
    const int r    = isB ? row - B * T : row;
    const float* src = (isB ? b : a) + (size_t)r * D + lane * 16;
    float*       dst = (isB ? nb : na) + (size_t)r * D + lane * 16;

    const float4* s4 = reinterpret_cast<const float4*>(src);
    float4 v[4];
    float ss = 0.f;
    #pragma unroll
    for (int q = 0; q < 4; ++q) {
        v[q] = s4[q];
        ss = fmaf(v[q].x, v[q].x, ss);
        ss = fmaf(v[q].y, v[q].y, ss);
        ss = fmaf(v[q].z, v[q].z, ss);
        ss = fmaf(v[q].w, v[q].w, ss);
    }
    ss = waveReduceSum(ss);
    const float sc = rsqrtf(fmaxf(ss, EPS));
    float4* d4 = reinterpret_cast<float4*>(dst);
    #pragma unroll
    for (int q = 0; q < 4; ++q) {
        float4 o = v[q];
        o.x *= sc; o.y *= sc; o.z *= sc; o.w *= sc;
        d4[q] = o;
    }
}

// ---------------------------------------------------------------------------
// Kernel 2: alpha[b] = nb[b]^T @ na[b]   (D x D, K = T = 256), fp32 WMMA.
// 2x2 register-blocked: one 32x32 output tile per wave (4 accumulators).
// A fragments reused across both N-tiles, B fragments across both M-tiles:
// 8 b32 loads/lane -> 4 WMMAs per k-step (vs 4 loads -> 1 WMMA unblocked).
// A layout (16x4 f32): lanes 0-15 hold M, VGPR0/1 = K,K+1; lanes 16-31 = K+2,K+3.
// ---------------------------------------------------------------------------
__global__ __launch_bounds__(128) void alpha_wmma(const float* __restrict__ na,
                                                  const float* __restrict__ nb,
                                                  float* __restrict__ alpha) {
    const int wave = threadIdx.x >> 5;
    const int lane = threadIdx.x & 31;
    const int half = lane >> 4;
    const int lid  = lane & 15;
    const int gid  = blockIdx.x * 4 + wave;           // 0 .. B*16*16-1
    const int b    = gid >> 8;                        // 256 macro-tiles per batch
    const int tix  = gid & 255;
    const int m0   = (tix >> 4) << 5;                 // 32-wide over D
    const int n0   = (tix & 15) << 5;                 // 32-wide over D

    const float* NB = nb + (size_t)b * T * D;
    const float* NA = na + (size_t)b * T * D;

    v8f c00 = {0.f,0.f,0.f,0.f,0.f,0.f,0.f,0.f};
    v8f c01 = c00, c10 = c00, c11 = c00;

    for (int k = 0; k < T; k += 4) {
        const int kk = k + 2 * half;
        const size_t r0 = (size_t)kk * D;
        const size_t r1 = (size_t)(kk + 1) * D;
        v2f a0, a1, b0, b1;
        a0.x = NB[r0 + m0 + lid];       a0.y = NB[r1 + m0 + lid];
        a1.x = NB[r0 + m0 + 16 + lid];  a1.y = NB[r1 + m0 + 16 + lid];
        b0.x = NA[r0 + n0 + lid];       b0.y = NA[r1 + n0 + lid];
        b1.x = NA[r0 + n0 + 16 + lid];  b1.y = NA[r1 + n0 + 16 + lid];
        c00 = __builtin_amdgcn_wmma_f32_16x16x4_f32(false, a0, false, b0, (short)0, c00, false, false);
        c01 = __builtin_amdgcn_wmma_f32_16x16x4_f32(false, a0, false, b1, (short)0, c01, false, false);
        c10 = __builtin_amdgcn_wmma_f32_16x16x4_f32(false, a1, false, b0, (short)0, c10, false, false);
        c11 = __builtin_amdgcn_wmma_f32_16x16x4_f32(false, a1, false, b1, (short)0, c11, false, false);
    }

    float* out = alpha + (size_t)b * D * D;
    #pragma unroll
    for (int r = 0; r < 8; ++r) {
        const size_t row0 = (size_t)(m0 + r + 8 * half) * D;
        const size_t row1 = (size_t)(m0 + 16 + r + 8 * half) * D;
        out[row0 + n0 + lid]      = c00[r];
        out[row0 + n0 + 16 + lid] = c01[r];
        out[row1 + n0 + lid]      = c10[r];
        out[row1 + n0 + 16 + lid] = c11[r];
    }
}

// ---------------------------------------------------------------------------
// Kernel 3: rcn[b,e] = rsqrt(max(sum_d alpha[b,d,e]^2, EPS)) — coalesced column walk.
// ---------------------------------------------------------------------------
__global__ __launch_bounds__(256) void col_norm(const float* __restrict__ alpha,
                                                float* __restrict__ rcn) {
    const int idx = blockIdx.x * 256 + threadIdx.x;   // 0 .. B*D-1
    const int b = idx / D;
    const int e = idx % D;
    const float* col = alpha + (size_t)b * D * D + e;
    float ss = 0.f;
    for (int d = 0; d < D; ++d) {
        const float v = col[(size_t)d * D];
        ss = fmaf(v, v, ss);
    }
    rcn[idx] = rsqrtf(fmaxf(ss, EPS));
}

// ---------------------------------------------------------------------------
// Kernel 4: hmean[b] = nb[b] @ alpha[b]  (T x D, K = D = 512), column-scaled by rcn.
// 2x2 register-blocked 32x32 tiles. A (nb) rows contiguous in K: float2 per lane.
// ---------------------------------------------------------------------------
__global__ __launch_bounds__(128) void hmean_wmma(const float* __restrict__ nb,
                                                  const float* __restrict__ alpha,
                                                  const float* __restrict__ rcn,
                                                  float* __restrict__ hm) {
    const int wave = threadIdx.x >> 5;
    const int lane = threadIdx.x & 31;
    const int half = lane >> 4;
    const int lid  = lane & 15;
    const int gid  = blockIdx.x * 4 + wave;           // 0 .. B*8*16-1
    const int b    = gid >> 7;                        // 128 macro-tiles per batch
    const int tix  = gid & 127;
    const int m0   = (tix >> 4) << 5;                 // 32-wide over T
    const int n0   = (tix & 15) << 5;                 // 32-wide over D

    const float* NB = nb + (size_t)b * T * D;
    const float* AL = alpha + (size_t)b * D * D;

    v8f c00 = {0.f,0.f,0.f,0.f,0.f,0.f,0.f,0.f};
    v8f c01 = c00, c10 = c00, c11 = c00;

    for (int k = 0; k < D; k += 4) {
        const int kk = k + 2 * half;
        const float2 ap0 = *reinterpret_cast<const float2*>(NB + (size_t)(m0 + lid) * D + kk);
        const float2 ap1 = *reinterpret_cast<const float2*>(NB + (size_t)(m0 + 16 + lid) * D + kk);
        const size_t r0 = (size_t)kk * D;
        const size_t r1 = (size_t)(kk + 1) * D;
        v2f a0, a1, b0, b1;
        a0.x = ap0.x;  a0.y = ap0.y;
        a1.x = ap1.x;  a1.y = ap1.y;
        b0.x = AL[r0 + n0 + lid];       b0.y = AL[r1 + n0 + lid];
        b1.x = AL[r0 + n0 + 16 + lid];  b1.y = AL[r1 + n0 + 16 + lid];
        c00 = __builtin_amdgcn_wmma_f32_16x16x4_f32(false, a0, false, b0, (short)0, c00, false, false);
        c01 = __builtin_amdgcn_wmma_f32_16x16x4_f32(false, a0, false, b1, (short)0, c01, false, false);
        c10 = __builtin_amdgcn_wmma_f32_16x16x4_f32(false, a1, false, b0, (short)0, c10, false, false);
        c11 = __builtin_amdgcn_wmma_f32_16x16x4_f32(false, a1, false, b1, (short)0, c11, false, false);
    }

    const float sc0 = rcn[b * D + n0 + lid];
    const float sc1 = rcn[b * D + n0 + 16 + lid];
    float* out = hm + (size_t)b * T * D;
    #pragma unroll
    for (int r = 0; r < 8; ++r) {
        const size_t row0 = (size_t)(m0 + r + 8 * half) * D;
        const size_t row1 = (size_t)(m0 + 16 + r + 8 * half) * D;
        out[row0 + n0 + lid]      = c00[r] * sc0;
        out[row0 + n0 + 16 + lid] = c01[r] * sc1;
        out[row1 + n0 + lid]      = c10[r] * sc0;
        out[row1 + n0 + 16 + lid] = c11[r] * sc1;
    }
}

// ---------------------------------------------------------------------------
// Kernel 5: perspectives. W staged in LDS (40 KB / block). One wave per (b,t) row.
// Strided lane partition d = lane + 32*j: coalesced global loads, conflict-free LDS.
// persp = s1 * rsqrt(max(s2,eps)) * rsqrt(max(s3,eps)); duplicated to both outputs.
// ---------------------------------------------------------------------------
__global__ __launch_bounds__(256) void persp_kernel(const float* __restrict__ inp_a,
                                                    const float* __restrict__ hmean,
                                                    const float* __restrict__ W,
                                                    float* __restrict__ out) {
    __shared__ float sW[P * D];
    for (int i = threadIdx.x; i < P * D; i += 256)
        sW[i] = W[i];
    __syncthreads();

    const int wave = threadIdx.x >> 5;
    const int lane = threadIdx.x & 31;
    const int row  = blockIdx.x * 8 + wave;           // 0 .. B*T-1

    const float* ap = inp_a + (size_t)row * D;
    const float* hp = hmean + (size_t)row * D;
    float av[16], hv[16];
    #pragma unroll
    for (int j = 0; j < 16; ++j) {
        av[j] = ap[lane + 32 * j];
        hv[j] = hp[lane + 32 * j];
    }

    for (int p = 0; p < P; ++p) {
        float s1 = 0.f, s2 = 0.f, s3 = 0.f;
        const float* wp = sW + p * D;
        #pragma unroll
        for (int j = 0; j < 16; ++j) {
            const float w  = wp[lane + 32 * j];
            const float ww = w * w;
            s1 = fmaf(av[j] * hv[j], ww, s1);
            s2 = fmaf(av[j] * av[j], ww, s2);
            s3 = fmaf(hv[j] * hv[j], ww, s3);
        }
        s1 = waveReduceSum(s1);
        s2 = waveReduceSum(s2);
        s3 = waveReduceSum(s3);
        if (lane == 0) {
            const float res = s1 * rsqrtf(fmaxf(s2, EPS)) * rsqrtf(fmaxf(s3, EPS));
            out[(size_t)row * P + p] = res;
            out[(size_t)B * T * P + (size_t)row * P + p] = res;   // tuple duplicate
        }
    }
}

// ---------------------------------------------------------------------------
// Launch pipeline.  Workspace layout (floats):
//   na[B*T*D] | nb[B*T*D] | alpha[B*D*D] | rcn[B*D]        (~67 MB total)
//   hmean reuses na's buffer (na dead after alpha_wmma; step 5 uses raw inp_a).
// ---------------------------------------------------------------------------
extern "C" void kernel_launch(void* const* d_in, const int* in_sizes, int n_in,
                              void* d_out, int out_size, void* d_ws, size_t ws_size,
                              hipStream_t stream) {
    const float* inp_a = (const float*)d_in[0];
    const float* inp_b = (const float*)d_in[1];
    const float* W     = (const float*)d_in[2];
    float* out = (float*)d_out;

    float* ws    = (float*)d_ws;
    float* na    = ws;
    float* nb    = na + (size_t)B * T * D;
    float* alpha = nb + (size_t)B * T * D;
    float* rcn   = alpha + (size_t)B * D * D;
    float* hmean = na;  // reuse

    normalize_rows<<<(2 * B * T) / 8, 256, 0, stream>>>(inp_a, inp_b, na, nb);
    alpha_wmma<<<(B * 16 * 16) / 4, 128, 0, stream>>>(na, nb, alpha);
    col_norm<<<(B * D) / 256, 256, 0, stream>>>(alpha, rcn);
    hmean_wmma<<<(B * 8 * 16) / 4, 128, 0, stream>>>(nb, alpha, rcn, hmean);
    persp_kernel<<<(B * T) / 8, 256, 0, stream>>>(inp_a, hmean, W, out);
}